// MultiScaleAttention_68521908240839
// MI455X (gfx1250) — compile-verified
//
#include <hip/hip_runtime.h>

#define H_    12
#define DK_   64
#define D_    768
#define B_    16
#define N_    512
#define KNN_  153      // int(0.3 * 512)
#define ROWS_ 8192     // B_*N_

typedef float v2f __attribute__((ext_vector_type(2)));
typedef float v8f __attribute__((ext_vector_type(8)));

__device__ __forceinline__ v8f vzero8() {
  v8f z = {0.f,0.f,0.f,0.f,0.f,0.f,0.f,0.f};
  return z;
}

// CDNA5 fp32 matrix op: D = A(16x4) * B(4x16) + C(16x16), wave32.
__device__ __forceinline__ v8f wmma_f32_k4(v2f a, v2f b, v8f c) {
  return __builtin_amdgcn_wmma_f32_16x16x4_f32(
      /*neg_a=*/false, a, /*neg_b=*/false, b,
      /*c_mod=*/(short)0, c, /*reuse_a=*/false, /*reuse_b=*/false);
}

// ---------------------------------------------------------------------------
// Kernel 1: Q/K/V projection.  out = x @ W^T + b, stored head-major [B,H,N,64].
// grid = (ROWS_/16, 3), block = 256 (8 waves). Each wave: 6 col-tiles of 16.
// ---------------------------------------------------------------------------
__global__ void __launch_bounds__(256)
qkv_proj_kernel(const float* __restrict__ x,
                const float* __restrict__ qw, const float* __restrict__ qb,
                const float* __restrict__ kw, const float* __restrict__ kb,
                const float* __restrict__ vw, const float* __restrict__ vb,
                float* __restrict__ Q, float* __restrict__ K, float* __restrict__ V)
{
  __shared__ float sA[16 * 772];   // 16x768 x-tile, padded stride (bank spread)
  const int which = blockIdx.y;
  const float* W    = (which == 0) ? qw : (which == 1) ? kw : vw;
  const float* bias = (which == 0) ? qb : (which == 1) ? kb : vb;
  float* dst        = (which == 0) ? Q  : (which == 1) ? K  : V;

  const int r0 = blockIdx.x * 16;
  for (int i = threadIdx.x; i < 3072; i += 256) {      // 16*768/4 float4 loads
    int row = i / 192, c4 = i % 192;
    float4 t4 = *(const float4*)(x + (size_t)(r0 + row) * D_ + c4 * 4);
    float* dr = &sA[row * 772 + c4 * 4];
    dr[0] = t4.x; dr[1] = t4.y; dr[2] = t4.z; dr[3] = t4.w;
  }
  __syncthreads();

  const int lane = threadIdx.x & 31, wave = threadIdx.x >> 5;
  const int l15 = lane & 15, khalf = lane >> 4;
  const int c0 = wave * 96;

  v8f acc[6];
  #pragma unroll
  for (int t = 0; t < 6; ++t) acc[t] = vzero8();

  #pragma unroll 1
  for (int kk = 0; kk < D_; kk += 4) {
    // A frag: lanes 0-15 hold (M=lane, K=kk..kk+1); lanes 16-31 K=kk+2..kk+3
    v2f a = *(const v2f*)&sA[l15 * 772 + kk + 2 * khalf];
    #pragma unroll
    for (int t = 0; t < 6; ++t) {
      int n = c0 + t * 16 + l15;                       // output column = W row
      v2f b = *(const v2f*)(W + (size_t)n * D_ + kk + 2 * khalf);
      acc[t] = wmma_f32_k4(a, b, acc[t]);
    }
  }

  const int bb = r0 >> 9;                              // batch
  #pragma unroll
  for (int t = 0; t < 6; ++t) {
    int col = c0 + t * 16 + l15;
    int h = col >> 6, dk = col & 63;
    float bv = bias[col];
    #pragma unroll
    for (int i = 0; i < 8; ++i) {
      int n = (r0 & 511) + i + 8 * khalf;
      dst[(((size_t)bb * H_ + h) * N_ + n) * DK_ + dk] = acc[t][i] + bv;
    }
  }
}

// ---------------------------------------------------------------------------
// Kernel 2: kNN mask.  One wave per (b,row): binary-search the 153rd-smallest
// distance threshold (distances are uniform [0,1)), emit 512-bit row mask.
// grid = ROWS_/8, block = 256 (8 waves).
// ---------------------------------------------------------------------------
__global__ void __launch_bounds__(256)
knn_mask_kernel(const float* __restrict__ dist, unsigned* __restrict__ maskw)
{
  const int lane = threadIdx.x & 31, wave = threadIdx.x >> 5;
  const int row = blockIdx.x * 8 + wave;               // 0..8191
  const int b = row >> 9, i = row & 511;

  float vals[16];
  #pragma unroll
  for (int u = 0; u < 16; ++u) {
    int j = lane * 16 + u;
    vals[u] = (i == 0 || j == 0) ? 0.f
            : dist[((size_t)b * 511 + (i - 1)) * 511 + (j - 1)];
  }

  float lo = 0.f, hi = 1.f;
  for (int it = 0; it < 24; ++it) {
    float mid = 0.5f * (lo + hi);
    int c = 0;
    #pragma unroll
    for (int u = 0; u < 16; ++u) c += (vals[u] <= mid) ? 1 : 0;
    for (int off = 16; off; off >>= 1) c += __shfl_xor(c, off, 32);
    if (c >= KNN_) hi = mid; else lo = mid;
  }

  unsigned m = 0;
  #pragma unroll
  for (int u = 0; u < 16; ++u) {
    int j = lane * 16 + u;
    if ((vals[u] <= hi) || (i == 0) || (j == 0)) m |= (1u << u);
  }
  unsigned partner = __shfl_xor(m, 1, 32);
  if ((lane & 1) == 0)
    maskw[(size_t)row * 16 + (lane >> 1)] = m | (partner << 16);
}

// ---------------------------------------------------------------------------
// Kernel 3a: scores = QK^T/8, kNN mask, softmax -> attn (written once).
// grid = B*H*(N/16), block = 256. Each wave: 16 rows x 64 cols, K=64.
// ---------------------------------------------------------------------------
__global__ void __launch_bounds__(256)
attn_softmax_kernel(const float* __restrict__ Q, const float* __restrict__ K,
                    const unsigned* __restrict__ maskw, float* __restrict__ attn)
{
  __shared__ float sS[16 * 512];                       // 32 KB score strip
  const int blk = blockIdx.x;
  const int bh = blk >> 5, rblk = blk & 31;
  const int b = bh / H_;
  const int lane = threadIdx.x & 31, wave = threadIdx.x >> 5;
  const int l15 = lane & 15, khalf = lane >> 4;
  const int r0 = rblk * 16;
  const float* Qp = Q + ((size_t)bh * N_ + r0) * DK_;
  const float* Kp = K + (size_t)bh * N_ * DK_;
  const int c0 = wave * 64;

  v8f acc[4];
  #pragma unroll
  for (int t = 0; t < 4; ++t) acc[t] = vzero8();

  #pragma unroll 1
  for (int kk = 0; kk < DK_; kk += 4) {
    v2f a = *(const v2f*)(Qp + (size_t)l15 * DK_ + kk + 2 * khalf);
    #pragma unroll
    for (int t = 0; t < 4; ++t) {
      int n = c0 + t * 16 + l15;
      v2f bb2 = *(const v2f*)(Kp + (size_t)n * DK_ + kk + 2 * khalf);
      acc[t] = wmma_f32_k4(a, bb2, acc[t]);
    }
  }

  const float scale = 0.125f;                          // 1/sqrt(64)
  #pragma unroll
  for (int t = 0; t < 4; ++t) {
    int col = c0 + t * 16 + l15;
    #pragma unroll
    for (int i = 0; i < 8; ++i) {
      int lrow = i + 8 * khalf;
      unsigned w = maskw[((size_t)b * N_ + (r0 + lrow)) * 16 + (col >> 5)];
      bool on = (w >> (col & 31)) & 1u;
      sS[lrow * 512 + col] = on ? acc[t][i] * scale : -1e12f;
    }
  }
  __syncthreads();

  // softmax: 2 rows per wave; cols strided by 32 -> conflict-free, coalesced
  #pragma unroll
  for (int rr = 0; rr < 2; ++rr) {
    int lrow = wave * 2 + rr;
    float v[16];
    float mx = -3.4e38f;
    #pragma unroll
    for (int e = 0; e < 16; ++e) {
      v[e] = sS[lrow * 512 + lane + 32 * e];
      mx = fmaxf(mx, v[e]);
    }
    for (int off = 16; off; off >>= 1) mx = fmaxf(mx, __shfl_xor(mx, off, 32));
    float sum = 0.f;
    #pragma unroll
    for (int e = 0; e < 16; ++e) { v[e] = __expf(v[e] - mx); sum += v[e]; }
    for (int off = 16; off; off >>= 1) sum += __shfl_xor(sum, off, 32);
    float inv = 1.f / sum;
    float* dst = attn + ((size_t)bh * N_ + (r0 + lrow)) * N_;
    #pragma unroll
    for (int e = 0; e < 16; ++e) dst[lane + 32 * e] = v[e] * inv;
  }
}

// ---------------------------------------------------------------------------
// Kernel 3b: msg = attn @ V, stored concat-head [B,N,768] (transpose is free).
// grid = B*H*(N/16), block = 128 (4 waves, one 16x16 dk-tile each, K=512).
// ---------------------------------------------------------------------------
__global__ void __launch_bounds__(128)
msg_kernel(const float* __restrict__ attn, const float* __restrict__ V,
           float* __restrict__ msg)
{
  __shared__ float sA[16 * 516];                       // attn tile, padded
  const int blk = blockIdx.x;
  const int bh = blk >> 5, rblk = blk & 31;
  const int b = bh / H_, h = bh % H_;
  const int r0 = rblk * 16;
  const float* Ap = attn + ((size_t)bh * N_ + r0) * N_;

  for (int i = threadIdx.x; i < 16 * 128; i += 128) {  // 2048 float4 loads
    int row = i >> 7, c4 = i & 127;
    float4 t4 = *(const float4*)(Ap + (size_t)row * N_ + c4 * 4);
    float* dr = &sA[row * 516 + c4 * 4];
    dr[0] = t4.x; dr[1] = t4.y; dr[2] = t4.z; dr[3] = t4.w;
  }
  __syncthreads();

  const int lane = threadIdx.x & 31, wave = threadIdx.x >> 5;
  const int l15 = lane & 15, khalf = lane >> 4;
  const int c0 = wave * 16;                            // dk tile base
  const float* Vp = V + (size_t)bh * N_ * DK_;

  v8f acc = vzero8();
  #pragma unroll 1
  for (int kk = 0; kk < N_; kk += 4) {
    v2f a = *(const v2f*)&sA[l15 * 516 + kk + 2 * khalf];
    int kB = kk + 2 * khalf;
    v2f bb2 = { Vp[(size_t)kB * DK_ + c0 + l15],
                Vp[(size_t)(kB + 1) * DK_ + c0 + l15] };
    acc = wmma_f32_k4(a, bb2, acc);
  }

  #pragma unroll
  for (int i = 0; i < 8; ++i) {
    int n = r0 + i + 8 * khalf;
    int col = h * DK_ + c0 + l15;
    msg[((size_t)b * N_ + n) * D_ + col] = acc[i];
  }
}

// ---------------------------------------------------------------------------
// Kernel 4a: y1 = silu(msg @ s1^T + s1b).  Same GEMM template as kernel 1.
// ---------------------------------------------------------------------------
__global__ void __launch_bounds__(256)
mlp1_kernel(const float* __restrict__ msg, const float* __restrict__ s1w,
            const float* __restrict__ s1b, float* __restrict__ y1)
{
  __shared__ float sA[16 * 772];
  const int r0 = blockIdx.x * 16;
  for (int i = threadIdx.x; i < 3072; i += 256) {
    int row = i / 192, c4 = i % 192;
    float4 t4 = *(const float4*)(msg + (size_t)(r0 + row) * D_ + c4 * 4);
    float* dr = &sA[row * 772 + c4 * 4];
    dr[0] = t4.x; dr[1] = t4.y; dr[2] = t4.z; dr[3] = t4.w;
  }
  __syncthreads();

  const int lane = threadIdx.x & 31, wave = threadIdx.x >> 5;
  const int l15 = lane & 15, khalf = lane >> 4;
  const int c0 = wave * 96;

  v8f acc[6];
  #pragma unroll
  for (int t = 0; t < 6; ++t) acc[t] = vzero8();

  #pragma unroll 1
  for (int kk = 0; kk < D_; kk += 4) {
    v2f a = *(const v2f*)&sA[l15 * 772 + kk + 2 * khalf];
    #pragma unroll
    for (int t = 0; t < 6; ++t) {
      int n = c0 + t * 16 + l15;
      v2f bb2 = *(const v2f*)(s1w + (size_t)n * D_ + kk + 2 * khalf);
      acc[t] = wmma_f32_k4(a, bb2, acc[t]);
    }
  }

  #pragma unroll
  for (int t = 0; t < 6; ++t) {
    int col = c0 + t * 16 + l15;
    float bv = s1b[col];
    #pragma unroll
    for (int i = 0; i < 8; ++i) {
      float vv = acc[t][i] + bv;
      float sv = vv * (1.f / (1.f + __expf(-vv)));     // SiLU
      y1[(size_t)(r0 + i + 8 * khalf) * D_ + col] = sv;
    }
  }
}

// ---------------------------------------------------------------------------
// Kernel 4b: out = LN(y1 @ s2^T + s2b + x) * g + b.  GEMM + fused epilogue.
// ---------------------------------------------------------------------------
__global__ void __launch_bounds__(256)
mlp2_ln_kernel(const float* __restrict__ y1, const float* __restrict__ s2w,
               const float* __restrict__ s2b, const float* __restrict__ x,
               const float* __restrict__ ln_g, const float* __restrict__ ln_b,
               float* __restrict__ out)
{
  __shared__ float sA[16 * 772];
  __shared__ float red1[256], red2[256];
  __shared__ float smu[16], srs[16];
  const int r0 = blockIdx.x * 16;

  for (int i = threadIdx.x; i < 3072; i += 256) {
    int row = i / 192, c4 = i % 192;
    float4 t4 = *(const float4*)(y1 + (size_t)(r0 + row) * D_ + c4 * 4);
    float* dr = &sA[row * 772 + c4 * 4];
    dr[0] = t4.x; dr[1] = t4.y; dr[2] = t4.z; dr[3] = t4.w;
  }
  __syncthreads();

  const int lane = threadIdx.x & 31, wave = threadIdx.x >> 5;
  const int l15 = lane & 15, khalf = lane >> 4;
  const int c0 = wave * 96;

  v8f acc[6];
  #pragma unroll
  for (int t = 0; t < 6; ++t) acc[t] = vzero8();

  #pragma unroll 1
  for (int kk = 0; kk < D_; kk += 4) {
    v2f a = *(const v2f*)&sA[l15 * 772 + kk + 2 * khalf];
    #pragma unroll
    for (int t = 0; t < 6; ++t) {
      int n = c0 + t * 16 + l15;
      v2f bb2 = *(const v2f*)(s2w + (size_t)n * D_ + kk + 2 * khalf);
      acc[t] = wmma_f32_k4(a, bb2, acc[t]);
    }
  }
  __syncthreads();                                     // done reading sA

  // Overwrite sA with pre-LN output tile: gemm + bias + residual.
  #pragma unroll
  for (int t = 0; t < 6; ++t) {
    int col = c0 + t * 16 + l15;
    float bv = s2b[col];
    #pragma unroll
    for (int i = 0; i < 8; ++i) {
      int lrow = i + 8 * khalf;
      sA[lrow * 772 + col] = acc[t][i] + bv + x[(size_t)(r0 + lrow) * D_ + col];
    }
  }
  __syncthreads();

  // Per-row mean / variance: 16 threads per row, 48 elements each.
  {
    int row = threadIdx.x >> 4, seg = threadIdx.x & 15;
    float s = 0.f, ss = 0.f;
    for (int j = seg * 48; j < seg * 48 + 48; ++j) {
      float vv = sA[row * 772 + j];
      s += vv; ss += vv * vv;
    }
    red1[threadIdx.x] = s; red2[threadIdx.x] = ss;
  }
  __syncthreads();
  if (threadIdx.x < 16) {
    float s = 0.f, ss = 0.f;
    for (int k = 0; k < 16; ++k) { s += red1[threadIdx.x * 16 + k]; ss += red2[threadIdx.x * 16 + k]; }
    float mu = s * (1.f / 768.f);
    float var = ss * (1.f / 768.f) - mu * mu;
    smu[threadIdx.x] = mu;
    srs[threadIdx.x] = rsqrtf(var + 1e-6f);
  }
  __syncthreads();

  for (int r = 0; r < 16; ++r) {
    float mu = smu[r], rs = srs[r];
    for (int col = threadIdx.x; col < D_; col += 256) {
      float vv = (sA[r * 772 + col] - mu) * rs * ln_g[col] + ln_b[col];
      out[(size_t)(r0 + r) * D_ + col] = vv;
    }
  }
}

// ---------------------------------------------------------------------------
extern "C" void kernel_launch(void* const* d_in, const int* in_sizes, int n_in,
                              void* d_out, int out_size, void* d_ws, size_t ws_size,
                              hipStream_t stream)
{
  (void)in_sizes; (void)n_in; (void)out_size; (void)ws_size;

  const float* x    = (const float*)d_in[0];
  const float* dist = (const float*)d_in[1];
  const float* q_w  = (const float*)d_in[2];
  const float* q_b  = (const float*)d_in[3];
  const float* k_w  = (const float*)d_in[4];
  const float* k_b  = (const float*)d_in[5];
  const float* v_w  = (const float*)d_in[6];
  const float* v_b  = (const float*)d_in[7];
  const float* s1_w = (const float*)d_in[8];
  const float* s1_b = (const float*)d_in[9];
  const float* s2_w = (const float*)d_in[10];
  const float* s2_b = (const float*)d_in[11];
  const float* ln_g = (const float*)d_in[12];
  const float* ln_b = (const float*)d_in[13];

  float* out  = (float*)d_out;
  float* attn = out + (size_t)B_ * N_ * D_;            // [B,H,1,N,N] flat

  const size_t hsz = (size_t)B_ * H_ * N_ * DK_;       // 6.29M floats
  float* Qs  = (float*)d_ws;
  float* Ks  = Qs + hsz;
  float* Vs  = Ks + hsz;
  float* msg = Vs + hsz;
  float* y1  = msg + (size_t)ROWS_ * D_;
  unsigned* maskw = (unsigned*)(y1 + (size_t)ROWS_ * D_);

  qkv_proj_kernel<<<dim3(ROWS_ / 16, 3), 256, 0, stream>>>(
      x, q_w, q_b, k_w, k_b, v_w, v_b, Qs, Ks, Vs);
  knn_mask_kernel<<<dim3(ROWS_ / 8), 256, 0, stream>>>(dist, maskw);
  attn_softmax_kernel<<<dim3(B_ * H_ * (N_ / 16)), 256, 0, stream>>>(
      Qs, Ks, maskw, attn);
  msg_kernel<<<dim3(B_ * H_ * (N_ / 16)), 128, 0, stream>>>(attn, Vs, msg);
  mlp1_kernel<<<dim3(ROWS_ / 16), 256, 0, stream>>>(msg, s1_w, s1_b, y1);
  mlp2_ln_kernel<<<dim3(ROWS_ / 16), 256, 0, stream>>>(
      y1, s2_w, s2_b, x, ln_g, ln_b, out);
}